// BiBoTopkRouter_75239237091862
// MI455X (gfx1250) — compile-verified
//
#include <hip/hip_runtime.h>
#include <math.h>

// ---------------------------------------------------------------------------
// BiBo Top-K router for MI455X (gfx1250).
//
// Problem: logits = hidden[16384,2048] @ gate_w^T[2048,64]  (fp32),
//          then per-token top-8 + softmax.
// Roofline: 4.3 GFLOP vs 137 MiB moved -> HBM-bound (~5.9 us @ 23.3 TB/s).
// Therefore: keep fp32, use V_WMMA_F32_16X16X4_F32, maximize independent
// waves streaming hidden_states; gate_w (512 KiB) lives in WGP$/L2.
// ---------------------------------------------------------------------------

typedef __attribute__((ext_vector_type(2))) float v2f;
typedef __attribute__((ext_vector_type(8))) float v8f;

namespace {
constexpr int kH = 2048;          // hidden dim (K)
constexpr int kE = 64;            // experts (N)
constexpr int kTopK = 8;
constexpr int kTokens = 4 * 4096; // B*S = 16384 (M)
constexpr int kLogitElems = kTokens * kE;        // 1048576
constexpr int kWeightElems = kTokens * kTopK;    // 131072
}

// ---------------------------------------------------------------------------
// Kernel 1: router GEMM via WMMA f32 16x16x4.
// One wave handles one 16-token M-tile x all 4 N-tiles (64 experts).
// A fragment (16x4 f32): lanes 0-15 -> M=lane, K={k,k+1}; lanes 16-31 ->
// M=lane-16, K={k+2,k+3}  => one b64 load per lane per K-step.
// B fragment (4x16 f32): mirrored layout; B[k][n] = gate_w[n*H + k].
// ---------------------------------------------------------------------------
__global__ __launch_bounds__(64) void router_gemm_wmma(
    const float* __restrict__ hidden, const float* __restrict__ gate,
    float* __restrict__ logits) {
  const int lane = threadIdx.x & 31;
  const int wave = blockIdx.x * 2 + (threadIdx.x >> 5);
  const int tokBase = wave * 16;

  const int n = lane & 15;              // N / M sub-index within half-wave
  const int kOff = (lane >> 4) << 1;    // lanes 16-31 read K pair +2

  const float* aPtr  = hidden + (size_t)(tokBase + n) * kH + kOff;
  const float* bPtr0 = gate + (size_t)(0 * 16 + n) * kH + kOff;
  const float* bPtr1 = gate + (size_t)(1 * 16 + n) * kH + kOff;
  const float* bPtr2 = gate + (size_t)(2 * 16 + n) * kH + kOff;
  const float* bPtr3 = gate + (size_t)(3 * 16 + n) * kH + kOff;

  v8f c0 = {}, c1 = {}, c2 = {}, c3 = {};

#pragma unroll 8
  for (int k = 0; k < kH; k += 4) {
    // Prefetch the hidden (HBM) stream ~2KB ahead, once per 128B line.
    // locality=3 -> WGP-scope prefetch: fills the near (WGP$) cache, not
    // just GL2, so the 8 consecutive K-steps that consume each line hit.
    if ((k & 28) == 0) {
      __builtin_prefetch(aPtr + k + 512, 0, 3);
    }
    v2f a  = *(const v2f*)(aPtr + k);
    v2f b0 = *(const v2f*)(bPtr0 + k);
    v2f b1 = *(const v2f*)(bPtr1 + k);
    v2f b2 = *(const v2f*)(bPtr2 + k);
    v2f b3 = *(const v2f*)(bPtr3 + k);
    // 8 args: (neg_a, A, neg_b, B, c_mod, C, reuse_a, reuse_b)
    c0 = __builtin_amdgcn_wmma_f32_16x16x4_f32(false, a, false, b0, (short)0, c0, false, false);
    c1 = __builtin_amdgcn_wmma_f32_16x16x4_f32(false, a, false, b1, (short)0, c1, false, false);
    c2 = __builtin_amdgcn_wmma_f32_16x16x4_f32(false, a, false, b2, (short)0, c2, false, false);
    c3 = __builtin_amdgcn_wmma_f32_16x16x4_f32(false, a, false, b3, (short)0, c3, false, false);
  }

  // C/D 16x16 f32 layout: lane -> N (lane&15), VGPR r -> M=r (lanes 0-15)
  // or M=r+8 (lanes 16-31). TEMP == 1.0 so logits are stored unscaled.
  const int mBase = (lane >> 4) * 8;
  float* o = logits + (size_t)(tokBase + mBase) * kE + n;
#pragma unroll
  for (int r = 0; r < 8; ++r) {
    float* row = o + (size_t)r * kE;
    row[0]  = c0[r];
    row[16] = c1[r];
    row[32] = c2[r];
    row[48] = c3[r];
  }
}

// ---------------------------------------------------------------------------
// Kernel 2: per-token top-8 (stable: descending value, ascending index, as
// jax.lax.top_k) + fp32 softmax over the selected 8. One thread per token.
// Selection is done lexicographically with no dynamic register indexing so
// the 64 logits stay entirely in VGPRs (no scratch).
// ---------------------------------------------------------------------------
__global__ __launch_bounds__(256) void router_topk_softmax(
    const float* __restrict__ logits, float* __restrict__ weights,
    int* __restrict__ experts) {
  const int tok = blockIdx.x * blockDim.x + threadIdx.x;
  if (tok >= kTokens) return;

  float v[kE];
  const float4* src = (const float4*)(logits + (size_t)tok * kE);
#pragma unroll
  for (int i = 0; i < kE / 4; ++i) {
    float4 t = src[i];
    v[4 * i + 0] = t.x;
    v[4 * i + 1] = t.y;
    v[4 * i + 2] = t.z;
    v[4 * i + 3] = t.w;
  }

  float wv[kTopK];
  int   wi[kTopK];
  float pv = INFINITY;   // previous selected value
  int   pi = -1;         // previous selected index
#pragma unroll
  for (int s = 0; s < kTopK; ++s) {
    float best = -INFINITY;
    int bi = 0;
#pragma unroll
    for (int e = 0; e < kE; ++e) {
      const float x = v[e];
      // eligible if strictly after (pv, pi) in (desc value, asc index) order
      const bool elig = (x < pv) || ((x == pv) && (e > pi));
      if (elig && (x > best)) { best = x; bi = e; }
    }
    wv[s] = best;
    wi[s] = bi;
    pv = best;
    pi = bi;
  }

  // Softmax over the 8 winners in fp32 (wv[0] is the max).
  const float m = wv[0];
  float sum = 0.0f;
#pragma unroll
  for (int s = 0; s < kTopK; ++s) {
    const float ex = expf(wv[s] - m);
    wv[s] = ex;
    sum += ex;
  }
  const float inv = 1.0f / sum;

  float* wOut = weights + (size_t)tok * kTopK;
  int*   eOut = experts + (size_t)tok * kTopK;
#pragma unroll
  for (int s = 0; s < kTopK; ++s) {
    wOut[s] = wv[s] * inv;
    eOut[s] = wi[s];
  }
}

// ---------------------------------------------------------------------------
// d_out layout (reference return order, flat):
//   [0, 1048576)            router_logits   f32
//   [1048576, 1179648)      routing_weights f32
//   [1179648, 1310720)      selected_experts i32 (bit-stored in same buffer)
// ---------------------------------------------------------------------------
extern "C" void kernel_launch(void* const* d_in, const int* in_sizes, int n_in,
                              void* d_out, int out_size, void* d_ws, size_t ws_size,
                              hipStream_t stream) {
  (void)in_sizes; (void)n_in; (void)out_size; (void)d_ws; (void)ws_size;
  const float* hidden = (const float*)d_in[0];
  const float* gate   = (const float*)d_in[1];
  float* out = (float*)d_out;

  float* logits  = out;
  float* weights = out + kLogitElems;
  int*   experts = (int*)(out + kLogitElems + kWeightElems);

  // 16384 tokens / 16 per wave = 1024 waves; 2 waves (64 thr) per block.
  router_gemm_wmma<<<dim3(kTokens / 32), dim3(64), 0, stream>>>(hidden, gate, logits);
  // One thread per token.
  router_topk_softmax<<<dim3(kTokens / 256), dim3(256), 0, stream>>>(logits, weights, experts);
}